// Network_2388001816887
// MI455X (gfx1250) — compile-verified
//
#include <hip/hip_runtime.h>
#include <math.h>

// ---------------------------------------------------------------------------
// GCN(1->2048) -> GCN(2048->1024) -> BN -> Linear(1024->256) -> LeakyReLU ->
// Linear(256->124) -> log_softmax, collapsed algebraically:
//   everything before LeakyReLU is affine in two per-node scalars
//   (agg2[n], aggN[n]); only the final 256->124 GEMM is a real matmul,
//   executed with V_WMMA_F32_16X16X4_F32 (f32 precision, wave32).
// ---------------------------------------------------------------------------

#define N_NODES 20000
#define N_EDGES 160000
#define KDIM 256
#define NCLS 124
#define CPAD 128

typedef __attribute__((ext_vector_type(2))) float v2f;
typedef __attribute__((ext_vector_type(8))) float v8f;

// workspace layout (float offsets)
#define OFF_DINV 0          // deg, then dinv in-place            [20000]
#define OFF_AGG1 20000      // scalar GCN1 aggregate              [20000]
#define OFF_AGG2 40000      // scalar GCN2 aggregate of agg1      [20000]
#define OFF_AGGN 60000      // scalar GCN2 aggregate of ones      [20000]
#define OFF_A    80000      // a = (W1@W2)*bnscale                [1024]
#define OFF_B    81024      // b = (b1@W2)*bnscale                [1024]
#define OFF_C    82048      // c = (b2-mean)*bnscale+beta         [1024]
#define OFF_P    83072      // p = a@lW1                          [256]
#define OFF_Q    83328      // q = b@lW1                          [256]
#define OFF_R    83584      // r = c@lW1 + lb1                    [256]
#define OFF_BT   83840      // lW2^T zero-padded to 128x256       [32768]
// total: 116608 floats = 466,432 bytes of d_ws

__global__ void k_init_deg(float* __restrict__ deg) {
  int i = blockIdx.x * blockDim.x + threadIdx.x;
  if (i < N_NODES) deg[i] = 1.0f;  // self loop
}

__global__ void k_deg_scatter(const int* __restrict__ dst, float* __restrict__ deg) {
  int e = blockIdx.x * blockDim.x + threadIdx.x;
  if (e < N_EDGES) atomicAdd(&deg[dst[e]], 1.0f);
}

__global__ void k_dinv_self(const float* __restrict__ x, float* __restrict__ dinv,
                            float* __restrict__ agg1) {
  int i = blockIdx.x * blockDim.x + threadIdx.x;
  if (i < N_NODES) {
    float dv = rsqrtf(dinv[i]);   // dinv currently holds deg (>=1 always)
    dinv[i] = dv;
    agg1[i] = dv * dv * x[i];     // self-loop contribution
  }
}

__global__ void k_scatter1(const int* __restrict__ src, const int* __restrict__ dst,
                           const float* __restrict__ x, const float* __restrict__ dinv,
                           float* __restrict__ agg1) {
  int e = blockIdx.x * blockDim.x + threadIdx.x;
  if (e < N_EDGES) {
    int s = src[e], d = dst[e];
    atomicAdd(&agg1[d], dinv[s] * dinv[d] * x[s]);
  }
}

__global__ void k_self2(const float* __restrict__ dinv, const float* __restrict__ agg1,
                        float* __restrict__ agg2, float* __restrict__ aggN) {
  int i = blockIdx.x * blockDim.x + threadIdx.x;
  if (i < N_NODES) {
    float dv2 = dinv[i] * dinv[i];
    agg2[i] = dv2 * agg1[i];
    aggN[i] = dv2;
  }
}

__global__ void k_scatter2(const int* __restrict__ src, const int* __restrict__ dst,
                           const float* __restrict__ dinv, const float* __restrict__ agg1,
                           float* __restrict__ agg2, float* __restrict__ aggN) {
  int e = blockIdx.x * blockDim.x + threadIdx.x;
  if (e < N_EDGES) {
    int s = src[e], d = dst[e];
    float norm = dinv[s] * dinv[d];
    atomicAdd(&agg2[d], norm * agg1[s]);
    atomicAdd(&aggN[d], norm);
  }
}

// a/b/c 1024-vectors: fold W1@W2, b1@W2 and BatchNorm
__global__ void k_abc(const float* __restrict__ W1, const float* __restrict__ b1,
                      const float* __restrict__ W2, const float* __restrict__ b2,
                      const float* __restrict__ gam, const float* __restrict__ bet,
                      const float* __restrict__ rmean, const float* __restrict__ rvar,
                      float* __restrict__ av, float* __restrict__ bv, float* __restrict__ cv) {
  int j = blockIdx.x * blockDim.x + threadIdx.x;
  if (j < 1024) {
    float u = 0.0f, v = 0.0f;
#pragma unroll 8
    for (int k = 0; k < 2048; ++k) {
      float w = W2[k * 1024 + j];       // coalesced across j
      u = fmaf(W1[k], w, u);
      v = fmaf(b1[k], w, v);
    }
    float scale = gam[j] * rsqrtf(rvar[j] + 1e-5f);
    av[j] = u * scale;
    bv[j] = v * scale;
    cv[j] = (b2[j] - rmean[j]) * scale + bet[j];
  }
}

// p/q/r 256-vectors: fold Linear(1024->256)
__global__ void k_pqr(const float* __restrict__ av, const float* __restrict__ bv,
                      const float* __restrict__ cv, const float* __restrict__ lW1,
                      const float* __restrict__ lb1,
                      float* __restrict__ pv, float* __restrict__ qv, float* __restrict__ rv) {
  int i = threadIdx.x;  // one block of 256
  float p = 0.0f, q = 0.0f, r = 0.0f;
#pragma unroll 8
  for (int j = 0; j < 1024; ++j) {
    float w = lW1[j * 256 + i];         // coalesced across i
    p = fmaf(av[j], w, p);
    q = fmaf(bv[j], w, q);
    r = fmaf(cv[j], w, r);
  }
  pv[i] = p;
  qv[i] = q;
  rv[i] = r + lb1[i];
}

// transpose lW2 [256,124] -> Bt [128,256] (classes padded with zeros)
__global__ void k_bt(const float* __restrict__ lW2, float* __restrict__ Bt) {
  int n = blockIdx.x;   // 0..127 class
  int k = threadIdx.x;  // 0..255
  Bt[n * KDIM + k] = (n < NCLS) ? lW2[k * NCLS + n] : 0.0f;
}

// ---------------------------------------------------------------------------
// Fused: h4 = agg2*p + aggN*q + r ; h5 = leaky(h4) ; logits = h5 @ lW2 + lb2 ;
// out = log_softmax(logits). One wave = 16 nodes x 128 (padded) classes.
// A fragments are synthesized in-register (no A memory traffic at all).
// ---------------------------------------------------------------------------
__global__ void __launch_bounds__(256) k_gemm_softmax(
    const float* __restrict__ agg2, const float* __restrict__ aggN,
    const float* __restrict__ pv, const float* __restrict__ qv,
    const float* __restrict__ rv, const float* __restrict__ Bt,
    const float* __restrict__ lb2, float* __restrict__ out) {
  const int wave = threadIdx.x >> 5;
  const int lane = threadIdx.x & 31;
  const int l15 = lane & 15;
  const int half = lane >> 4;
  const int mbase = (blockIdx.x * 8 + wave) * 16;

  // Per-lane node scalars (A-matrix rows: lanes 0-15 and 16-31 both map M=lane&15)
  const int nA = mbase + l15;
  const int ia = (nA < N_NODES) ? nA : 0;
  float g2 = agg2[ia];
  float gN = aggN[ia];
  if (nA >= N_NODES) { g2 = 0.0f; gN = 0.0f; }  // padded rows -> harmless values

  // lb2 per output column for each of the 8 N-tiles
  float bias[8];
#pragma unroll
  for (int nt = 0; nt < 8; ++nt) {
    int c = nt * 16 + l15;
    bias[nt] = (c < NCLS) ? lb2[c] : 0.0f;
  }

  v8f acc[8];
  const v8f vzero = {0, 0, 0, 0, 0, 0, 0, 0};
#pragma unroll
  for (int nt = 0; nt < 8; ++nt) acc[nt] = vzero;

  const int kkoff = half * 2;  // A/B layout: lanes 16-31 hold K+2,K+3
#pragma unroll 2
  for (int k0 = 0; k0 < KDIM; k0 += 4) {
    const int kk = k0 + kkoff;
    // Build A fragment on the fly: h5[node][kk], h5[node][kk+1]
    const float2 pp = *(const float2*)(pv + kk);
    const float2 qq = *(const float2*)(qv + kk);
    const float2 rr = *(const float2*)(rv + kk);
    float h0 = fmaf(g2, pp.x, fmaf(gN, qq.x, rr.x));
    float h1 = fmaf(g2, pp.y, fmaf(gN, qq.y, rr.y));
    v2f A;
    A.x = (h0 >= 0.0f) ? h0 : 0.01f * h0;  // LeakyReLU
    A.y = (h1 >= 0.0f) ? h1 : 0.01f * h1;
#pragma unroll
    for (int nt = 0; nt < 8; ++nt) {
      // B fragment: lane holds Bt[class = nt*16+l15][kk], [kk+1]
      const float2 bb = *(const float2*)(Bt + (nt * 16 + l15) * KDIM + kk);
      v2f B;
      B.x = bb.x;
      B.y = bb.y;
      acc[nt] = __builtin_amdgcn_wmma_f32_16x16x4_f32(
          false, A, false, B, (short)0, acc[nt], false, false);
    }
  }

  // log_softmax over 124 classes. C/D layout: VGPR j -> row j (lanes 0-15),
  // row 8+j (lanes 16-31); column = nt*16 + (lane&15). XOR shuffles with
  // masks <=8 stay within each 16-lane half, so both halves reduce their own
  // rows simultaneously.
#pragma unroll
  for (int j = 0; j < 8; ++j) {
    const int node = mbase + j + 8 * half;
    float vals[8];
    float vmax = -1e30f;
#pragma unroll
    for (int nt = 0; nt < 8; ++nt) {
      int c = nt * 16 + l15;
      float v = acc[nt][j] + bias[nt];
      if (c >= NCLS) v = -1e30f;  // padded classes never win max, exp -> 0
      vals[nt] = v;
      vmax = fmaxf(vmax, v);
    }
#pragma unroll
    for (int m = 8; m >= 1; m >>= 1) vmax = fmaxf(vmax, __shfl_xor(vmax, m, 32));
    float s = 0.0f;
#pragma unroll
    for (int nt = 0; nt < 8; ++nt) s += expf(vals[nt] - vmax);
#pragma unroll
    for (int m = 8; m >= 1; m >>= 1) s += __shfl_xor(s, m, 32);
    const float L = vmax + logf(s);
    if (node < N_NODES) {
#pragma unroll
      for (int nt = 0; nt < 8; ++nt) {
        int c = nt * 16 + l15;
        if (c < NCLS) out[node * NCLS + c] = vals[nt] - L;
      }
    }
  }
}

extern "C" void kernel_launch(void* const* d_in, const int* in_sizes, int n_in,
                              void* d_out, int out_size, void* d_ws, size_t ws_size,
                              hipStream_t stream) {
  const float* x = (const float*)d_in[0];
  const int* ei = (const int*)d_in[1];
  const float* W1 = (const float*)d_in[2];
  const float* b1 = (const float*)d_in[3];
  const float* W2 = (const float*)d_in[4];
  const float* b2 = (const float*)d_in[5];
  const float* gam = (const float*)d_in[6];
  const float* bet = (const float*)d_in[7];
  const float* rmean = (const float*)d_in[8];
  const float* rvar = (const float*)d_in[9];
  const float* lW1 = (const float*)d_in[10];
  const float* lb1 = (const float*)d_in[11];
  const float* lW2 = (const float*)d_in[12];
  const float* lb2 = (const float*)d_in[13];
  float* out = (float*)d_out;
  float* ws = (float*)d_ws;

  const int* src = ei;             // edge_index[0]
  const int* dst = ei + N_EDGES;   // edge_index[1]

  const int nb_nodes = (N_NODES + 255) / 256;
  const int nb_edges = (N_EDGES + 255) / 256;

  k_init_deg<<<nb_nodes, 256, 0, stream>>>(ws + OFF_DINV);
  k_deg_scatter<<<nb_edges, 256, 0, stream>>>(dst, ws + OFF_DINV);
  k_dinv_self<<<nb_nodes, 256, 0, stream>>>(x, ws + OFF_DINV, ws + OFF_AGG1);
  k_scatter1<<<nb_edges, 256, 0, stream>>>(src, dst, x, ws + OFF_DINV, ws + OFF_AGG1);
  k_self2<<<nb_nodes, 256, 0, stream>>>(ws + OFF_DINV, ws + OFF_AGG1, ws + OFF_AGG2,
                                        ws + OFF_AGGN);
  k_scatter2<<<nb_edges, 256, 0, stream>>>(src, dst, ws + OFF_DINV, ws + OFF_AGG1,
                                           ws + OFF_AGG2, ws + OFF_AGGN);
  k_abc<<<4, 256, 0, stream>>>(W1, b1, W2, b2, gam, bet, rmean, rvar,
                               ws + OFF_A, ws + OFF_B, ws + OFF_C);
  k_pqr<<<1, 256, 0, stream>>>(ws + OFF_A, ws + OFF_B, ws + OFF_C, lW1, lb1,
                               ws + OFF_P, ws + OFF_Q, ws + OFF_R);
  k_bt<<<CPAD, KDIM, 0, stream>>>(lW2, ws + OFF_BT);

  const int mblocks = (N_NODES + 127) / 128;  // 8 waves x 16 nodes per block
  k_gemm_softmax<<<mblocks, 256, 0, stream>>>(ws + OFF_AGG2, ws + OFF_AGGN,
                                              ws + OFF_P, ws + OFF_Q, ws + OFF_R,
                                              ws + OFF_BT, lb2, out);
}